// Model_45191645889359
// MI455X (gfx1250) — compile-verified
//
#include <hip/hip_runtime.h>
#include <stdint.h>

// fp32 RMS-norm chains, MI455X (gfx1250, wave32).
// Bandwidth-bound: ~667MB traffic -> ~29us floor @ 23.3 TB/s.
// z branch (90% of traffic) stages through LDS via CDNA5 async-to-LDS ops so
// every global read is fully coalesced (512B contiguous per wave instruction).

typedef float v4f __attribute__((ext_vector_type(4)));

#define EPS_DEF 1.1920928955078125e-07f  // torch fp32 eps (2^-23)

__device__ __forceinline__ float rsq(float x) { return __builtin_amdgcn_rsqf(x); }

// full-wave (32 lane) sum; valid result in lane 0
__device__ __forceinline__ float wave_sum(float v) {
#pragma unroll
  for (int off = 16; off > 0; off >>= 1) v += __shfl_down(v, off, 32);
  return v;
}

#if defined(__gfx1250__) && __has_builtin(__builtin_amdgcn_global_load_async_to_lds_b128)
#define USE_ASYNC_LDS 1
#else
#define USE_ASYNC_LDS 0
#endif

#if USE_ASYNC_LDS
// builtin signature (from compiler diagnostic): param0 = AS1 (global) int4*,
// param1 = AS3 (LDS) int4*, then imm offset, imm cpol.
typedef int v4i_mem __attribute__((vector_size(16)));
typedef __attribute__((address_space(1))) v4i_mem* as1_p4;
typedef __attribute__((address_space(3))) v4i_mem* as3_p4;
// generic->AS1: identical 64-bit representation for global pointers
__device__ __forceinline__ as1_p4 as_global4(const void* p) {
  return (as1_p4)(unsigned long long)(uintptr_t)p;
}
// generic->AS3: low 32 bits of a generic LDS pointer are the LDS offset
__device__ __forceinline__ as3_p4 as_lds4(const void* p) {
  return (as3_p4)(unsigned int)(uintptr_t)p;
}
__device__ __forceinline__ void wait_async0() {
#if __has_builtin(__builtin_amdgcn_s_wait_asynccnt)
  __builtin_amdgcn_s_wait_asynccnt(0);
#else
  asm volatile("s_wait_asynccnt 0x0" ::: "memory");
#endif
}
#endif

// ---------------- x branch: norm(24,w0) -> norm(12x24) -> norm(24,w3) -------
// block = 192 threads = 16 slabs of 12 rows; thread owns one row of 24 floats.
__global__ void __launch_bounds__(192) rms_x_kernel(
    const float* __restrict__ x, const float* __restrict__ w0,
    const float* __restrict__ w3, float* __restrict__ out) {
  __shared__ float s_ss[192];
  __shared__ float s_w0[24];
  __shared__ float s_w3[24];
  const int tid = threadIdx.x;
  if (tid < 24) { s_w0[tid] = w0[tid]; s_w3[tid] = w3[tid]; }

  const long long row = (long long)blockIdx.x * 192 + tid;
  const v4f* p = (const v4f*)(x + row * 24);
  v4f a[6];
#pragma unroll
  for (int k = 0; k < 6; ++k) a[k] = __builtin_nontemporal_load(p + k);

  float ss0 = 0.f;
#pragma unroll
  for (int k = 0; k < 6; ++k)
    ss0 += a[k][0]*a[k][0] + a[k][1]*a[k][1] + a[k][2]*a[k][2] + a[k][3]*a[k][3];
  const float r1 = rsq(ss0 * (1.0f/24.0f) + EPS_DEF);

  __syncthreads();  // weights visible

  // ss1 = sum((x*r1*w0)^2) = r1^2 * sum((x*w0)^2)
  float sw = 0.f;
#pragma unroll
  for (int j = 0; j < 24; ++j) {
    const float t = a[j >> 2][j & 3] * s_w0[j];
    sw += t * t;
  }
  const float ss1 = r1 * r1 * sw;

  // reduce ss1 over the 12 rows of this slab (threads [12g, 12g+12))
  s_ss[tid] = ss1;
  __syncthreads();
  const int base = (tid / 12) * 12;
  float S = 0.f;
#pragma unroll
  for (int k = 0; k < 12; ++k) S += s_ss[base + k];

  const float r2 = rsq(S * (1.0f/288.0f) + EPS_DEF);
  const float r3 = rsq(r2 * r2 * ss1 * (1.0f/24.0f) + EPS_DEF);
  const float c = r1 * r2 * r3;

  v4f* q = (v4f*)(out + row * 24);
#pragma unroll
  for (int k = 0; k < 6; ++k) {
    v4f o;
#pragma unroll
    for (int e = 0; e < 4; ++e) {
      const int j = 4 * k + e;
      o[e] = a[k][e] * c * s_w0[j] * s_w3[j];
    }
    __builtin_nontemporal_store(o, q + k);
  }
}

// ---- y branch: norm(16,eps=1e-3) -> norm(12x16,w1) -> norm(12x16,w4) -------
// Both (12,16) norms fuse into ONE reduction of (sum y1^2, sum (y1*w1)^2).
__global__ void __launch_bounds__(192) rms_y_kernel(
    const float* __restrict__ y, const float* __restrict__ w1,
    const float* __restrict__ w4, float* __restrict__ out) {
  __shared__ float s_w1[192];
  __shared__ float s_w4[192];
  __shared__ float s_a[192];
  __shared__ float s_b[192];
  const int tid = threadIdx.x;
  s_w1[tid] = w1[tid];
  s_w4[tid] = w4[tid];

  const long long row = (long long)blockIdx.x * 192 + tid;
  const v4f* p = (const v4f*)(y + row * 16);
  v4f a[4];
#pragma unroll
  for (int k = 0; k < 4; ++k) a[k] = __builtin_nontemporal_load(p + k);

  float ss0 = 0.f;
#pragma unroll
  for (int k = 0; k < 4; ++k)
    ss0 += a[k][0]*a[k][0] + a[k][1]*a[k][1] + a[k][2]*a[k][2] + a[k][3]*a[k][3];
  const float r1 = rsq(ss0 * (1.0f/16.0f) + 1.0e-3f);

  __syncthreads();  // weights visible

  const int wb = (tid % 12) * 16;
  float swB = 0.f;
#pragma unroll
  for (int j = 0; j < 16; ++j) {
    const float t = a[j >> 2][j & 3] * s_w1[wb + j];
    swB += t * t;
  }
  const float ssA = r1 * r1 * ss0;
  const float ssB = r1 * r1 * swB;

  s_a[tid] = ssA;
  s_b[tid] = ssB;
  __syncthreads();
  const int base = (tid / 12) * 12;
  float SA = 0.f, SB = 0.f;
#pragma unroll
  for (int k = 0; k < 12; ++k) { SA += s_a[base + k]; SB += s_b[base + k]; }

  const float r2 = rsq(SA * (1.0f/192.0f) + EPS_DEF);
  const float r3 = rsq(r2 * r2 * SB * (1.0f/192.0f) + EPS_DEF);
  const float c = r1 * r2 * r3;

  v4f* q = (v4f*)(out + row * 16);
#pragma unroll
  for (int k = 0; k < 4; ++k) {
    v4f o;
#pragma unroll
    for (int e = 0; e < 4; ++e) {
      const int j = 4 * k + e;
      o[e] = a[k][e] * c * s_w1[wb + j] * s_w4[wb + j];
    }
    __builtin_nontemporal_store(o, q + k);
  }
}

// -- z branch: norm(24,w2) -> norm(12x16x24,eps=1e-2) -> norm(24,w5) ---------
// block = 192 threads = one (b,s) slab of 192 rows x 24 (18 KB).
// Stage slab through LDS with CDNA5 async-to-LDS: contiguous global reads,
// padded rows in LDS (stride 7 x float4 = 112B) to dodge bank conflicts.
__global__ void __launch_bounds__(192) rms_z_kernel(
    const float* __restrict__ z, const float* __restrict__ w2,
    const float* __restrict__ w5, float* __restrict__ out) {
  __shared__ float s_w2[24];
  __shared__ float s_w5[24];
  __shared__ float s_part[6];
  const int tid = threadIdx.x;
  if (tid < 24) { s_w2[tid] = w2[tid]; s_w5[tid] = w5[tid]; }

  const long long row = (long long)blockIdx.x * 192 + tid;
  v4f a[6];

#if USE_ASYNC_LDS
  __shared__ v4f buf4[192 * 7];  // 192 rows, 6 data float4 + 1 pad each
  const char* gsrc = (const char*)(z + (long long)blockIdx.x * 4608);
#pragma unroll
  for (int k = 0; k < 6; ++k) {
    const int f = tid + k * 192;   // float4 chunk index in slab: contiguous/wave
    const int r = f / 6;           // destination row
    const int e = f - r * 6;       // float4 within row
    __builtin_amdgcn_global_load_async_to_lds_b128(
        as_global4(gsrc + (long long)f * 16), as_lds4(&buf4[r * 7 + e]),
        /*offset=*/0, /*cpol=*/0);
  }
  wait_async0();       // this wave's async writes to LDS complete
  __syncthreads();     // all waves' writes visible (also covers weights)
#pragma unroll
  for (int k = 0; k < 6; ++k) a[k] = buf4[tid * 7 + k];
#else
  const v4f* p = (const v4f*)(z + row * 24);
#pragma unroll
  for (int k = 0; k < 6; ++k) a[k] = __builtin_nontemporal_load(p + k);
  __syncthreads();     // weights visible
#endif

  float ss0 = 0.f;
#pragma unroll
  for (int k = 0; k < 6; ++k)
    ss0 += a[k][0]*a[k][0] + a[k][1]*a[k][1] + a[k][2]*a[k][2] + a[k][3]*a[k][3];
  const float r1 = rsq(ss0 * (1.0f/24.0f) + EPS_DEF);

  float sw = 0.f;
#pragma unroll
  for (int j = 0; j < 24; ++j) {
    const float t = a[j >> 2][j & 3] * s_w2[j];
    sw += t * t;
  }
  const float ss1 = r1 * r1 * sw;

  // slab-wide reduction over 4608 elems: wave32 shuffle + 6 LDS partials
  const float S = wave_sum(ss1);
  if ((tid & 31) == 0) s_part[tid >> 5] = S;
  __syncthreads();
  float T = 0.f;
#pragma unroll
  for (int k = 0; k < 6; ++k) T += s_part[k];

  const float r2 = rsq(T * (1.0f/4608.0f) + 1.0e-2f);
  const float r3 = rsq(r2 * r2 * ss1 * (1.0f/24.0f) + EPS_DEF);
  const float c = r1 * r2 * r3;

  v4f* q = (v4f*)(out + row * 24);
#pragma unroll
  for (int k = 0; k < 6; ++k) {
    v4f o;
#pragma unroll
    for (int e = 0; e < 4; ++e) {
      const int j = 4 * k + e;
      o[e] = a[k][e] * c * s_w2[j] * s_w5[j];
    }
    __builtin_nontemporal_store(o, q + k);
  }
}

extern "C" void kernel_launch(void* const* d_in, const int* in_sizes, int n_in,
                              void* d_out, int out_size, void* d_ws, size_t ws_size,
                              hipStream_t stream) {
  const float* x  = (const float*)d_in[0];
  const float* y  = (const float*)d_in[1];
  const float* z  = (const float*)d_in[2];
  const float* w0 = (const float*)d_in[3];
  const float* w1 = (const float*)d_in[4];
  const float* w2 = (const float*)d_in[5];
  const float* w3 = (const float*)d_in[6];
  const float* w4 = (const float*)d_in[7];
  const float* w5 = (const float*)d_in[8];

  const int nx = in_sizes[0];  // 32*512*12*24    = 4718592
  const int ny = in_sizes[1];  // 32*512*12*16    = 3145728
  const int nz = in_sizes[2];  // 32*512*12*16*24 = 75497472

  float* outx = (float*)d_out;
  float* outy = outx + nx;
  float* outz = outy + ny;

  const dim3 blk(192);
  rms_x_kernel<<<nx / (24 * 192), blk, 0, stream>>>(x, w0, w3, outx);  // 1024 blocks
  rms_y_kernel<<<ny / (16 * 192), blk, 0, stream>>>(y, w1, w4, outy);  // 1024 blocks
  rms_z_kernel<<<nz / (24 * 192), blk, 0, stream>>>(z, w2, w5, outz);  // 16384 blocks
}